// GraphEncoder_7507602833648
// MI455X (gfx1250) — compile-verified
//
#include <hip/hip_runtime.h>
#include <math.h>

typedef float v2f __attribute__((ext_vector_type(2)));
typedef float v8f __attribute__((ext_vector_type(8)));
typedef unsigned long long u64;

#if defined(__gfx1250__) && __has_builtin(__builtin_amdgcn_wmma_f32_16x16x4_f32)
#define USE_WMMA_F32X4 1
#endif

// D = A(16x4) * B(4x16) + C, fp32 WMMA. Layouts per CDNA5 ISA 7.12.2:
//  A: lane<16 -> M=lane, {v0,v1}={K0,K1}; lane>=16 -> M=lane-16, {K2,K3}
//  B: lane<16 -> N=lane, {v0,v1}={K0,K1}; lane>=16 -> N=lane-16, {K2,K3}
//  C/D: vgpr r -> M = r + 8*(lane/16), N = lane%16
__device__ __forceinline__ v8f wmma4_f32(v2f a, v2f b, v8f c, int lane) {
#if defined(USE_WMMA_F32X4)
  (void)lane;
  return __builtin_amdgcn_wmma_f32_16x16x4_f32(false, a, false, b, (short)0, c,
                                               false, false);
#else
  int half = lane >> 4, row = lane & 15;
#pragma unroll
  for (int r = 0; r < 8; ++r) {
    int m = r + 8 * half;
    float sum = 0.f;
#pragma unroll
    for (int K = 0; K < 4; ++K) {
      float av = __shfl((K & 1) ? a.y : a.x, ((K >> 1) << 4) + m, 32);
      float bv = __shfl((K & 1) ? b.y : b.x, ((K >> 1) << 4) + row, 32);
      sum += av * bv;
    }
    c[r] += sum;
  }
  return c;
#endif
}

// ---------------- utility kernels ----------------
__global__ void k_fill_f32(float* p, float v, int n) {
  int i = blockIdx.x * blockDim.x + threadIdx.x;
  if (i < n) p[i] = v;
}
__global__ void k_fill_i32(int* p, int v, int n) {
  int i = blockIdx.x * blockDim.x + threadIdx.x;
  if (i < n) p[i] = v;
}
// src/dst/ew init in one pass
__global__ void k_init_edges(int* src, int* dst, float* ew, const int* eidx, int E) {
  int e = blockIdx.x * blockDim.x + threadIdx.x;
  if (e >= E) return;
  src[e] = eidx[e];
  dst[e] = eidx[E + e];
  ew[e] = 1.0f;
}

// ---------------- degree / normalization ----------------
__global__ void k_deg(const int* __restrict__ dst, const float* __restrict__ ew,
                      float* deg, int E) {
  int e = blockIdx.x * blockDim.x + threadIdx.x;
  if (e >= E) return;
  float w = ew ? ew[e] : 1.0f;
  if (w != 0.f) atomicAdd(&deg[dst[e]], w);
}
__global__ void k_deg_to_dinv(float* d, int n) {
  int i = blockIdx.x * blockDim.x + threadIdx.x;
  if (i < n) d[i] = rsqrtf(d[i] + 1.0f);  // +1 self loop, always >= 1
}

// ---------------- GEMM 1: hproj[n,16] = x[n,256] @ W[256,16] (WMMA f32) -----
__global__ void k_gemm_in(const float* __restrict__ x, const float* __restrict__ W,
                          float* __restrict__ hp, int n) {
  __shared__ float sW[256 * 16];
  {
    const float4* W4 = (const float4*)W;
    float4* sW4 = (float4*)sW;
    for (int i = threadIdx.x; i < 256 * 16 / 4; i += blockDim.x) sW4[i] = W4[i];
  }
  __syncthreads();
  int lane = threadIdx.x & 31;
  int tile = blockIdx.x * (blockDim.x >> 5) + (threadIdx.x >> 5);
  int m0 = tile << 4;
  if (m0 >= n) return;  // wave-uniform -> EXEC all ones inside
  int half = lane >> 4, row = lane & 15;
  int mrow = m0 + row;
  if (mrow > n - 1) mrow = n - 1;  // clamp: OOB rows compute garbage, never stored
  const v2f* xr = (const v2f*)(x + (size_t)mrow * 256);
  v8f acc = {};
#pragma unroll
  for (int k = 0; k < 256; k += 4) {
    v2f a = xr[(k >> 1) + half];  // unconditional global_load_b64
    int ka = k + (half << 1);
    v2f b;
    b.x = sW[ka * 16 + row];
    b.y = sW[ka * 16 + 16 + row];
    acc = wmma4_f32(a, b, acc, lane);
  }
#pragma unroll
  for (int r = 0; r < 8; ++r) {
    int m = m0 + r + (half << 3);
    if (m < n) hp[(size_t)m * 16 + row] = acc[r];
  }
}

// ---------------- 16-channel edge aggregation ----------------
__global__ void k_agg16(const int* __restrict__ src, const int* __restrict__ dst,
                        const float* __restrict__ ew, const float* __restrict__ dinv,
                        const float* __restrict__ h, float* agg, int E) {
  int t = blockIdx.x * blockDim.x + threadIdx.x;
  int e = t >> 4, c = t & 15;
  if (e >= E) return;
  int s = src[e], d = dst[e];
  float w = ew ? ew[e] : 1.0f;
  float coef = dinv[s] * w * dinv[d];
  if (coef != 0.f) atomicAdd(&agg[(size_t)d * 16 + c], h[(size_t)s * 16 + c] * coef);
}

// h_out = (relu?)(agg + hp*dinv^2 + bias)
__global__ void k_finish16(const float* __restrict__ agg, const float* __restrict__ hp,
                           const float* __restrict__ dinv,
                           const float* __restrict__ bias, float* h, int n, int relu) {
  int t = blockIdx.x * blockDim.x + threadIdx.x;
  int i = t >> 4, c = t & 15;
  if (i >= n) return;
  float di = dinv[i];
  float v = agg[(size_t)i * 16 + c] + hp[(size_t)i * 16 + c] * di * di +
            (bias ? bias[c] : 0.f);
  h[(size_t)i * 16 + c] = relu ? fmaxf(v, 0.f) : v;
}

// ---------------- pooling score path ----------------
__global__ void k_sproj(const float* __restrict__ h, const float* __restrict__ Wp,
                        float* sp, int n) {
  int i = blockIdx.x * blockDim.x + threadIdx.x;
  if (i >= n) return;
  const float* hr = h + (size_t)i * 16;
  float acc = 0.f;
#pragma unroll
  for (int c = 0; c < 16; ++c) acc += hr[c] * Wp[c];
  sp[i] = acc;
}
__global__ void k_sagg(const int* __restrict__ src, const int* __restrict__ dst,
                       const float* __restrict__ ew, const float* __restrict__ dinv,
                       const float* __restrict__ sp, float* sagg, int E) {
  int e = blockIdx.x * blockDim.x + threadIdx.x;
  if (e >= E) return;
  int s = src[e], d = dst[e];
  float coef = dinv[s] * ew[e] * dinv[d];
  if (coef != 0.f) atomicAdd(&sagg[d], sp[s] * coef);
}
__global__ void k_sfinish(const float* sagg, const float* sp, const float* dinv,
                          const float* bp, float* score, int n) {
  int i = blockIdx.x * blockDim.x + threadIdx.x;
  if (i >= n) return;
  float di = dinv[i];
  score[i] = tanhf(sagg[i] + sp[i] * di * di + bp[0]);
}

// ---------------- top-k via bitonic sort of packed u64 keys ----------------
// key = (order-preserving(score) << 32) | (~index); overall DESCENDING sort,
// so equal scores tie-break toward smaller index (matches jax.lax.top_k).
__global__ void k_pack_keys(const float* score, u64* keys, int n, int P) {
  int i = blockIdx.x * blockDim.x + threadIdx.x;
  if (i >= P) return;
  u64 kv = 0ull;  // pad keys sort last (all real keys > 0)
  if (i < n) {
    unsigned u = __float_as_uint(score[i]);
    u ^= (u >> 31) ? 0xFFFFFFFFu : 0x80000000u;
    kv = ((u64)u << 32) | (unsigned)(0xFFFFFFFFu - (unsigned)i);
  }
  keys[i] = kv;
}

#define CHUNK 2048  // keys per LDS-resident block (1024 threads x 2)

// all stages k=2..CHUNK fused in LDS (direction from GLOBAL index)
__global__ __launch_bounds__(1024) void k_bsort_local(u64* keys) {
  __shared__ u64 s[CHUNK];
  int base = blockIdx.x * CHUNK;
  s[threadIdx.x] = keys[base + threadIdx.x];
  s[threadIdx.x + 1024] = keys[base + threadIdx.x + 1024];
  __syncthreads();
  for (int k = 2; k <= CHUNK; k <<= 1) {
    for (int j = k >> 1; j > 0; j >>= 1) {
      int t = threadIdx.x;
      int i = ((t & ~(j - 1)) << 1) | (t & (j - 1));
      int p = i | j;
      bool desc = (((base + i) & k) == 0);
      u64 a = s[i], b = s[p];
      bool sw = desc ? (a < b) : (a > b);
      if (sw) { s[i] = b; s[p] = a; }
      __syncthreads();
    }
  }
  keys[base + threadIdx.x] = s[threadIdx.x];
  keys[base + threadIdx.x + 1024] = s[threadIdx.x + 1024];
}

// for a given outer k (> CHUNK): all stages j = CHUNK/2 .. 1 fused in LDS
__global__ __launch_bounds__(1024) void k_bmerge_local(u64* keys, int k) {
  __shared__ u64 s[CHUNK];
  int base = blockIdx.x * CHUNK;
  s[threadIdx.x] = keys[base + threadIdx.x];
  s[threadIdx.x + 1024] = keys[base + threadIdx.x + 1024];
  __syncthreads();
  bool desc = ((base & k) == 0);  // k > CHUNK, i < CHUNK -> uniform per block
  for (int j = CHUNK >> 1; j > 0; j >>= 1) {
    int t = threadIdx.x;
    int i = ((t & ~(j - 1)) << 1) | (t & (j - 1));
    int p = i | j;
    u64 a = s[i], b = s[p];
    bool sw = desc ? (a < b) : (a > b);
    if (sw) { s[i] = b; s[p] = a; }
    __syncthreads();
  }
  keys[base + threadIdx.x] = s[threadIdx.x];
  keys[base + threadIdx.x + 1024] = s[threadIdx.x + 1024];
}

// global step, only used while j >= CHUNK
__global__ void k_bitonic(u64* keys, int j, int k, int P) {
  int i = blockIdx.x * blockDim.x + threadIdx.x;
  if (i >= P) return;
  int ixj = i ^ j;
  if (ixj > i) {
    u64 a = keys[i], b = keys[ixj];
    bool desc = ((i & k) == 0);
    bool sw = desc ? (a < b) : (a > b);
    if (sw) { keys[i] = b; keys[ixj] = a; }
  }
}

__global__ void k_extract(const u64* keys, int* perm, float* vals, int k) {
  int i = blockIdx.x * blockDim.x + threadIdx.x;
  if (i >= k) return;
  u64 kv = keys[i];
  unsigned u = (unsigned)(kv >> 32);
  u ^= (u >> 31) ? 0x80000000u : 0xFFFFFFFFu;
  vals[i] = __uint_as_float(u);
  perm[i] = (int)(0xFFFFFFFFu - (unsigned)(kv & 0xFFFFFFFFull));
}

__global__ void k_gather16(const float* __restrict__ h, const int* __restrict__ perm,
                           const float* __restrict__ vals, float* o, int k) {
  int t = blockIdx.x * blockDim.x + threadIdx.x;
  int i = t >> 4, c = t & 15;
  if (i >= k) return;
  o[(size_t)i * 16 + c] = h[(size_t)perm[i] * 16 + c] * vals[i];
}
__global__ void k_scatter_inv(const int* perm, int* inv, int k) {
  int i = blockIdx.x * blockDim.x + threadIdx.x;
  if (i < k) inv[perm[i]] = i;
}
__global__ void k_relabel(int* src, int* dst, float* ew, const int* inv, int E) {
  int e = blockIdx.x * blockDim.x + threadIdx.x;
  if (e >= E) return;
  int ns = inv[src[e]], nd = inv[dst[e]];
  bool valid = (ns >= 0) && (nd >= 0);
  ew[e] = valid ? ew[e] : 0.f;
  src[e] = valid ? ns : 0;
  dst[e] = valid ? nd : 0;
}

// -------- GEMM 2: out[n,128] = relu(z[n,16] @ W[16,128] + b) (WMMA f32) ----
__global__ void k_gemm_out(const float* __restrict__ z, const float* __restrict__ W,
                           const float* __restrict__ bias, float* __restrict__ out,
                           int n) {
  __shared__ float sW[16 * 128];
  {
    const float4* W4 = (const float4*)W;
    float4* sW4 = (float4*)sW;
    for (int i = threadIdx.x; i < 16 * 128 / 4; i += blockDim.x) sW4[i] = W4[i];
  }
  __syncthreads();
  int lane = threadIdx.x & 31;
  int gt = blockIdx.x * (blockDim.x >> 5) + (threadIdx.x >> 5);
  int mt = gt >> 3, nt = gt & 7;
  int m0 = mt << 4, n0 = nt << 4;
  if (m0 >= n) return;
  int half = lane >> 4, row = lane & 15;
  int mrow = m0 + row;
  if (mrow > n - 1) mrow = n - 1;  // clamp (see k_gemm_in)
  const v2f* zr = (const v2f*)(z + (size_t)mrow * 16);
  v8f acc = {};
#pragma unroll
  for (int k = 0; k < 16; k += 4) {
    v2f a = zr[(k >> 1) + half];
    int ka = k + (half << 1);
    v2f b;
    b.x = sW[ka * 128 + n0 + row];
    b.y = sW[ka * 128 + 128 + n0 + row];
    acc = wmma4_f32(a, b, acc, lane);
  }
  float bb = bias[n0 + row];
#pragma unroll
  for (int r = 0; r < 8; ++r) {
    int m = m0 + r + (half << 3);
    if (m < n) out[(size_t)m * 128 + n0 + row] = fmaxf(acc[r] + bb, 0.f);
  }
}

// ---------------- host orchestration ----------------
static inline int cdiv(int a, int b) { return (a + b - 1) / b; }

extern "C" void kernel_launch(void* const* d_in, const int* in_sizes, int n_in,
                              void* d_out, int out_size, void* d_ws, size_t ws_size,
                              hipStream_t stream) {
  const float* x     = (const float*)d_in[0];
  const int*   eidx  = (const int*)d_in[1];
  const int*   oidx  = (const int*)d_in[2];
  const float* W_in  = (const float*)d_in[3];
  const float* b_in  = (const float*)d_in[4];
  const float* Wp0   = (const float*)d_in[5];
  const float* bp0   = (const float*)d_in[6];
  const float* Wp1   = (const float*)d_in[7];
  const float* bp1   = (const float*)d_in[8];
  const float* W_out = (const float*)d_in[9];
  const float* b_out = (const float*)d_in[10];
  float* out = (float*)d_out;

  const int N  = in_sizes[0] / 256;  // 100000
  const int E  = in_sizes[1] / 2;    // 3200000
  const int OE = in_sizes[2] / 2;    // 800000
  const int K1 = (N + 1) / 2;        // 50000
  const int K2 = (K1 + 1) / 2;       // 25000
  int P1 = 1; while (P1 < N) P1 <<= 1;   // 131072
  int P2 = 1; while (P2 < K1) P2 <<= 1;  // 65536

  // workspace carve-out (256B aligned)
  size_t off = 0;
  char* base = (char*)d_ws;
  auto A = [&](size_t bytes) -> char* {
    char* p = base + off;
    off += (bytes + 255) & ~(size_t)255;
    return p;
  };
  int*   src   = (int*)A((size_t)E * 4);
  int*   dst   = (int*)A((size_t)E * 4);
  float* ew    = (float*)A((size_t)E * 4);
  float* hproj = (float*)A((size_t)N * 16 * 4);
  float* agg   = (float*)A((size_t)N * 16 * 4);
  float* hA    = (float*)A((size_t)N * 16 * 4);
  float* hB    = (float*)A((size_t)N * 16 * 4);
  float* dinv  = (float*)A((size_t)N * 4);
  float* sp    = (float*)A((size_t)N * 4);
  float* sagg  = (float*)A((size_t)N * 4);
  float* score = (float*)A((size_t)N * 4);
  u64*   keys  = (u64*)A((size_t)P1 * 8);
  int*   perm  = (int*)A((size_t)K1 * 4);
  float* vals  = (float*)A((size_t)K1 * 4);
  int*   inv   = (int*)A((size_t)N * 4);
  (void)ws_size; (void)n_in; (void)out_size;

  const int TB = 256;
  auto sort_topk = [&](int n, int P, int k) {
    k_pack_keys<<<cdiv(P, TB), TB, 0, stream>>>(score, keys, n, P);
    k_bsort_local<<<P / CHUNK, 1024, 0, stream>>>(keys);       // all k<=CHUNK
    for (int kk = CHUNK << 1; kk <= P; kk <<= 1) {
      for (int j = kk >> 1; j >= CHUNK; j >>= 1)               // cross-chunk steps
        k_bitonic<<<cdiv(P, TB), TB, 0, stream>>>(keys, j, kk, P);
      k_bmerge_local<<<P / CHUNK, 1024, 0, stream>>>(keys, kk); // j<CHUNK fused
    }
    k_extract<<<cdiv(k, TB), TB, 0, stream>>>(keys, perm, vals, k);
  };

  // ---- Stage 1: GCNConv 256->16 + ReLU (n=N, E edges, ew=1) ----
  k_init_edges<<<cdiv(E, TB), TB, 0, stream>>>(src, dst, ew, eidx, E);
  k_fill_f32<<<cdiv(N, TB), TB, 0, stream>>>(dinv, 0.f, N);
  k_deg<<<cdiv(E, TB), TB, 0, stream>>>(dst, ew, dinv, E);
  k_deg_to_dinv<<<cdiv(N, TB), TB, 0, stream>>>(dinv, N);
  k_gemm_in<<<cdiv(cdiv(N, 16), 4), 128, 0, stream>>>(x, W_in, hproj, N);
  k_fill_f32<<<cdiv(N * 16, TB), TB, 0, stream>>>(agg, 0.f, N * 16);
  k_agg16<<<cdiv(E * 16, TB), TB, 0, stream>>>(src, dst, ew, dinv, hproj, agg, E);
  k_finish16<<<cdiv(N * 16, TB), TB, 0, stream>>>(agg, hproj, dinv, b_in, hA, N, 1);

  // ---- Pool 0 (dinv reused: ew unchanged), hA[N] -> hB[K1] ----
  k_sproj<<<cdiv(N, TB), TB, 0, stream>>>(hA, Wp0, sp, N);
  k_fill_f32<<<cdiv(N, TB), TB, 0, stream>>>(sagg, 0.f, N);
  k_sagg<<<cdiv(E, TB), TB, 0, stream>>>(src, dst, ew, dinv, sp, sagg, E);
  k_sfinish<<<cdiv(N, TB), TB, 0, stream>>>(sagg, sp, dinv, bp0, score, N);
  sort_topk(N, P1, K1);
  k_gather16<<<cdiv(K1 * 16, TB), TB, 0, stream>>>(hA, perm, vals, hB, K1);
  k_fill_i32<<<cdiv(N, TB), TB, 0, stream>>>(inv, -1, N);
  k_scatter_inv<<<cdiv(K1, TB), TB, 0, stream>>>(perm, inv, K1);
  k_relabel<<<cdiv(E, TB), TB, 0, stream>>>(src, dst, ew, inv, E);

  // ---- Pool 1 (recompute deg with masked ew), hB[K1] -> hA[K2] ----
  k_fill_f32<<<cdiv(K1, TB), TB, 0, stream>>>(dinv, 0.f, K1);
  k_deg<<<cdiv(E, TB), TB, 0, stream>>>(dst, ew, dinv, E);
  k_deg_to_dinv<<<cdiv(K1, TB), TB, 0, stream>>>(dinv, K1);
  k_sproj<<<cdiv(K1, TB), TB, 0, stream>>>(hB, Wp1, sp, K1);
  k_fill_f32<<<cdiv(K1, TB), TB, 0, stream>>>(sagg, 0.f, K1);
  k_sagg<<<cdiv(E, TB), TB, 0, stream>>>(src, dst, ew, dinv, sp, sagg, E);
  k_sfinish<<<cdiv(K1, TB), TB, 0, stream>>>(sagg, sp, dinv, bp1, score, K1);
  sort_topk(K1, P2, K2);
  k_gather16<<<cdiv(K2 * 16, TB), TB, 0, stream>>>(hB, perm, vals, hA, K2);
  // (post-pool edge relabel skipped: the pooled graph is unused afterwards)

  // ---- Stage 3: GCNConv 16->128 over out_index; aggregate in 16-dim first
  //      (linearity: segsum(h@W * c) == segsum(h*c) @ W), then WMMA GEMM ----
  const int* osrc = oidx;
  const int* odst = oidx + OE;
  k_fill_f32<<<cdiv(K2, TB), TB, 0, stream>>>(dinv, 0.f, K2);
  k_deg<<<cdiv(OE, TB), TB, 0, stream>>>(odst, (const float*)nullptr, dinv, OE);
  k_deg_to_dinv<<<cdiv(K2, TB), TB, 0, stream>>>(dinv, K2);
  k_fill_f32<<<cdiv(K2 * 16, TB), TB, 0, stream>>>(agg, 0.f, K2 * 16);
  k_agg16<<<cdiv(OE * 16, TB), TB, 0, stream>>>(osrc, odst, (const float*)nullptr,
                                                dinv, hA, agg, OE);
  k_finish16<<<cdiv(K2 * 16, TB), TB, 0, stream>>>(agg, hA, dinv,
                                                   (const float*)nullptr, hB, K2, 0);
  k_gemm_out<<<cdiv(cdiv(K2, 16) * 8, 4), 128, 0, stream>>>(hB, W_out, b_out, out, K2);
}